// MultiScanVSSM_3702261809735
// MI455X (gfx1250) — compile-verified
//
#include <hip/hip_runtime.h>
#include <math.h>

#define B_    8
#define C_    256
#define H_    56
#define W_    56
#define L_    (H_ * W_)           // 3136
#define RED_  32
#define LN_EPS 1e-5f

typedef __attribute__((ext_vector_type(2))) float v2f;
typedef __attribute__((ext_vector_type(8))) float v8f;

// ---------------- workspace layout (floats) ----------------
// mu   : B*L            @ 0
// rstd : B*L            @ 25088
// S    : B*C            @ 50176
// T    : B              @ 52224
// gate : B*C            @ 52232
#define WS_MU    0
#define WS_RSTD  (B_ * L_)
#define WS_S     (2 * B_ * L_)
#define WS_T     (2 * B_ * L_ + B_ * C_)
#define WS_GATE  (2 * B_ * L_ + B_ * C_ + B_)

__device__ __forceinline__ float block_reduce_256(float v, float* red) {
    int tid = threadIdx.x;
    red[tid] = v;
    __syncthreads();
    for (int s = 128; s > 0; s >>= 1) {
        if (tid < s) red[tid] += red[tid + s];
        __syncthreads();
    }
    return red[0];
}

// ---------- Kernel 1: per-token mean / rstd over C (coalesced over L) ----------
__global__ void k_token_stats(const float* __restrict__ x, float* __restrict__ ws) {
    const int b   = blockIdx.y;
    const int l   = blockIdx.x * 256 + threadIdx.x;
    if (l >= L_) return;
    const float* xp = x + (size_t)b * C_ * L_ + l;
    float s = 0.f, s2 = 0.f;
#pragma unroll 8
    for (int c = 0; c < C_; ++c) {
        float v = xp[(size_t)c * L_];   // threads contiguous in l -> coalesced
        s  += v;
        s2 += v * v;
    }
    const float mu  = s * (1.0f / C_);
    const float var = s2 * (1.0f / C_) - mu * mu;
    ws[WS_MU   + b * L_ + l] = mu;
    ws[WS_RSTD + b * L_ + l] = rsqrtf(var + LN_EPS);
}

// ---------- Kernel 2: S[b,c] = sum_l rstd[b,l] * x[b,c,l] ----------
__global__ void k_channel_sums(const float* __restrict__ x, float* __restrict__ ws) {
    __shared__ float red[256];
    const int c = blockIdx.x;
    const int b = blockIdx.y;
    const float* xp = x + ((size_t)b * C_ + c) * L_;
    const float* rp = ws + WS_RSTD + b * L_;
    float p = 0.f;
    for (int l = threadIdx.x; l < L_; l += 256)
        p += rp[l] * xp[l];
    float tot = block_reduce_256(p, red);
    if (threadIdx.x == 0) ws[WS_S + b * C_ + c] = tot;
}

// ---------- Kernel 2b: T[b] = sum_l rstd[b,l] * mu[b,l] ----------
__global__ void k_token_dot(float* __restrict__ ws) {
    __shared__ float red[256];
    const int b = blockIdx.x;
    const float* mp = ws + WS_MU   + b * L_;
    const float* rp = ws + WS_RSTD + b * L_;
    float p = 0.f;
    for (int l = threadIdx.x; l < L_; l += 256)
        p += rp[l] * mp[l];
    float tot = block_reduce_256(p, red);
    if (threadIdx.x == 0) ws[WS_T + b] = tot;
}

__device__ __forceinline__ float gelu_exact(float v) {
    return 0.5f * v * (1.0f + erff(v * 0.70710678118654752f));
}
__device__ __forceinline__ float sigmoidf_(float v) {
    return 1.0f / (1.0f + expf(-v));
}

// ---------- Kernel 3: gate = sigmoid(gelu(G @ Wr + br) @ Ws + bs), WMMA f32 ----------
// G is [16 x 256] (rows 8..15 zero-padded), Wr [256 x 32], Ws [32 x 256].
// Single workgroup; wave 0 runs the V_WMMA_F32_16X16X4_F32 pipeline.
__global__ void k_gate(const float* __restrict__ ln_w, const float* __restrict__ ln_b,
                       const float* __restrict__ w_reduce, const float* __restrict__ b_reduce,
                       const float* __restrict__ w_select, const float* __restrict__ b_select,
                       float* __restrict__ ws) {
    __shared__ float gLds[16 * C_];    // 16 KB
    __shared__ float hLds[16 * RED_];  //  2 KB
    const int tid = threadIdx.x;

    // Build padded G: g[b,c] = ln_w[c]*(S-T)/L + ln_b[c]; rows >= 8 are zero.
    for (int idx = tid; idx < 16 * C_; idx += 256) {
        const int row = idx >> 8;       // /256
        const int c   = idx & 255;
        float v = 0.f;
        if (row < B_) {
            const float s = ws[WS_S + row * C_ + c];
            const float t = ws[WS_T + row];
            v = ln_w[c] * ((s - t) * (1.0f / L_)) + ln_b[c];
        }
        gLds[idx] = v;
    }
    __syncthreads();

    if (tid < 32) {                     // wave 0, EXEC all ones
        const int lane = tid;
        const int m    = lane & 15;     // row (A) / col (B,D)
        const int half = lane >> 4;
        const int kb   = half * 2;      // this lane's K sub-pair
        const int rowbase = half * 8;   // D rows held by this half-wave

        // GEMM1: H[16x32] = G[16x256] @ Wr[256x32]
        for (int nt = 0; nt < 2; ++nt) {
            v8f acc = {};
            const int ncol = nt * 16 + m;
#pragma unroll 4
            for (int k0 = 0; k0 < C_; k0 += 4) {
                v2f a, bmat;
                a.x    = gLds[m * C_ + k0 + kb];
                a.y    = gLds[m * C_ + k0 + kb + 1];
                bmat.x = w_reduce[(k0 + kb)     * RED_ + ncol];
                bmat.y = w_reduce[(k0 + kb + 1) * RED_ + ncol];
                acc = __builtin_amdgcn_wmma_f32_16x16x4_f32(
                        false, a, false, bmat, (short)0, acc, false, false);
            }
            const float bias = b_reduce[ncol];
#pragma unroll
            for (int j = 0; j < 8; ++j)
                hLds[(rowbase + j) * RED_ + ncol] = gelu_exact(acc[j] + bias);
        }
        // same-wave LDS store->load ordering handled by compiler waitcnt

        // GEMM2: Gate_pre[16x256] = H[16x32] @ Ws[32x256]
        for (int nt = 0; nt < 16; ++nt) {
            v8f acc = {};
            const int ncol = nt * 16 + m;
#pragma unroll
            for (int k0 = 0; k0 < RED_; k0 += 4) {
                v2f a, bmat;
                a.x    = hLds[m * RED_ + k0 + kb];
                a.y    = hLds[m * RED_ + k0 + kb + 1];
                bmat.x = w_select[(k0 + kb)     * C_ + ncol];
                bmat.y = w_select[(k0 + kb + 1) * C_ + ncol];
                acc = __builtin_amdgcn_wmma_f32_16x16x4_f32(
                        false, a, false, bmat, (short)0, acc, false, false);
            }
            if (half == 0) {            // D rows 0..7 are the real batches
                const float bias = b_select[ncol];
#pragma unroll
                for (int j = 0; j < 8; ++j)
                    ws[WS_GATE + j * C_ + ncol] = sigmoidf_(acc[j] + bias);
            }
        }
    }
}

// ---------- Kernel 4: out[b,l,c] = 8 * x[b,c,l] * gate[b,c]  (LDS transpose) ----------
__global__ void k_apply(const float* __restrict__ x, const float* __restrict__ ws,
                        float* __restrict__ out) {
    __shared__ float tile[32][33];
    const int lt = blockIdx.x;          // 98 l-tiles
    const int ct = blockIdx.y;          // 8 c-tiles
    const int b  = blockIdx.z;
    const int tx = threadIdx.x & 31;
    const int ty = threadIdx.x >> 5;    // 0..7
    const int l0 = lt * 32;
    const int c0 = ct * 32;

    const float* xp = x + ((size_t)b * C_) * L_;
#pragma unroll
    for (int i0 = 0; i0 < 32; i0 += 8) {
        const int cl = i0 + ty;
        tile[cl][tx] = xp[(size_t)(c0 + cl) * L_ + (l0 + tx)];
    }
    __syncthreads();

    const float gv = 8.0f * ws[WS_GATE + b * C_ + c0 + tx];
    float* op = out + ((size_t)b * L_) * C_;
#pragma unroll
    for (int j0 = 0; j0 < 32; j0 += 8) {
        const int ll = j0 + ty;
        op[(size_t)(l0 + ll) * C_ + (c0 + tx)] = tile[tx][ll] * gv;
    }
}

extern "C" void kernel_launch(void* const* d_in, const int* in_sizes, int n_in,
                              void* d_out, int out_size, void* d_ws, size_t ws_size,
                              hipStream_t stream) {
    const float* x        = (const float*)d_in[0];
    const float* ln_w     = (const float*)d_in[1];
    const float* ln_b     = (const float*)d_in[2];
    const float* w_reduce = (const float*)d_in[3];
    const float* b_reduce = (const float*)d_in[4];
    const float* w_select = (const float*)d_in[5];
    const float* b_select = (const float*)d_in[6];
    float* out = (float*)d_out;
    float* ws  = (float*)d_ws;

    (void)in_sizes; (void)n_in; (void)out_size; (void)ws_size;

    dim3 g1((L_ + 255) / 256, B_);            // 13 x 8
    k_token_stats<<<g1, 256, 0, stream>>>(x, ws);

    dim3 g2(C_, B_);                          // 256 x 8
    k_channel_sums<<<g2, 256, 0, stream>>>(x, ws);

    k_token_dot<<<B_, 256, 0, stream>>>(ws);

    k_gate<<<1, 256, 0, stream>>>(ln_w, ln_b, w_reduce, b_reduce,
                                  w_select, b_select, ws);

    dim3 g4(L_ / 32, C_ / 32, B_);            // 98 x 8 x 8
    k_apply<<<g4, 256, 0, stream>>>(x, ws, out);
}